// GCNModel_64493228917351
// MI455X (gfx1250) — compile-verified
//
#include <hip/hip_runtime.h>

typedef __attribute__((ext_vector_type(2))) float v2f;
typedef __attribute__((ext_vector_type(8))) float v8f;

#define GCN_IN_FEATS 512
#define GCN_HIDDEN   256
#define GCN_OUT      128

// ---------------------------------------------------------------------------
// Zero a float buffer, 16B vectorized (n must be a multiple of 4)
// ---------------------------------------------------------------------------
__global__ void zero_f32v4(float4* __restrict__ p, long long n4) {
  long long i = (long long)blockIdx.x * blockDim.x + threadIdx.x;
  long long stride = (long long)gridDim.x * blockDim.x;
  float4 z = make_float4(0.f, 0.f, 0.f, 0.f);
  for (; i < n4; i += stride) p[i] = z;
}

// ---------------------------------------------------------------------------
// Degree accumulation: deg_out[src[e]] += 1, deg_in[dst[e]] += 1
// ---------------------------------------------------------------------------
__global__ void degree_kernel(const int* __restrict__ src, const int* __restrict__ dst,
                              float* __restrict__ deg_out, float* __restrict__ deg_in,
                              int E) {
  int e = blockIdx.x * blockDim.x + threadIdx.x;
  if (e < E) {
    atomicAdd(&deg_out[src[e]], 1.0f);
    atomicAdd(&deg_in[dst[e]], 1.0f);
  }
}

// In-place: p[i] = rsqrt(max(p[i], 1))
__global__ void rsqrt_kernel(float* __restrict__ p, int n) {
  int i = blockIdx.x * blockDim.x + threadIdx.x;
  if (i < n) p[i] = rsqrtf(fmaxf(p[i], 1.0f));
}

// ---------------------------------------------------------------------------
// C[M,NC] = (diag(scale) * A[M,K]) @ B[K,NC]   (scale may be nullptr)
// f32 WMMA 16x16x4; one wave per 16x16 C tile; block = NC/16 waves covering a
// 16 x NC strip.  The scaled 16xK A strip is staged once per block in LDS,
// TRANSPOSED ([k][row]) so WMMA A-operand reads are bank-conflict-free:
// lanes 0-15 (K pair {k,k+1}) and lanes 16-31 (K pair {k+2,k+3}) read
// disjoint 16-bank spans of the 64-bank LDS.
//   A 16x4 f32 operand (ISA 7.12.2): lanes 0-15 -> M rows, VGPRs = K{0,1};
//   lanes 16-31 -> K{2,3}.  B 4x16 mirrored.  C/D: VGPR r ->
//   (M = r + 8*(lane>=16), N = lane&15).
// ---------------------------------------------------------------------------
template <int K, int NC>
__global__ __launch_bounds__(2 * NC)
void gemm_wmma_lds(const float* __restrict__ A,
                   const float* __restrict__ scale,
                   const float* __restrict__ B,
                   float* __restrict__ C) {
  __shared__ float lds_at[K * 16];          // [k][row], row-minor
  const int tid      = threadIdx.x;
  const int nthreads = 2 * NC;
  const int row_base = blockIdx.x * 16;

  // Cooperative, coalesced load of the 16xK strip, scale folded in, stored
  // transposed.  16*K is an exact multiple of nthreads -> no divergence,
  // EXEC stays all-ones for the WMMA loop below.
  #pragma unroll
  for (int g = tid; g < 16 * K; g += nthreads) {
    const int row = g / K;
    const int k   = g - row * K;
    const float s = scale ? scale[row_base + row] : 1.0f;
    lds_at[k * 16 + row] = A[(size_t)(row_base + row) * K + k] * s;
  }
  __syncthreads();

  const int lane = tid & 31;
  const int wave = tid >> 5;
  const int half = lane >> 4;               // 0: K pair {0,1}, 1: {2,3}
  const int l15  = lane & 15;

  const float* ap   = &lds_at[(half << 1) * 16 + l15];
  const float* bptr = B + (size_t)(half << 1) * NC + wave * 16 + l15;

  v8f acc = {0.f, 0.f, 0.f, 0.f, 0.f, 0.f, 0.f, 0.f};

  #pragma unroll 4
  for (int k = 0; k < K; k += 4) {
    v2f a, b;
    a.x = ap[0];                            // A[arow][k + 2*half]
    a.y = ap[16];                           // A[arow][k + 2*half + 1]
    b.x = bptr[0];                          // B[k + 2*half    ][col]
    b.y = bptr[NC];                         // B[k + 2*half + 1][col]
    acc = __builtin_amdgcn_wmma_f32_16x16x4_f32(
        /*neg_a=*/false, a, /*neg_b=*/false, b,
        /*c_mod=*/(short)0, acc, /*reuse_a=*/false, /*reuse_b=*/false);
    ap   += 64;                             // 4 k-slices * 16 rows
    bptr += (size_t)4 * NC;
  }

  float* cp = C + (size_t)(row_base + half * 8) * NC + wave * 16 + l15;
  #pragma unroll
  for (int r = 0; r < 8; ++r) cp[(size_t)r * NC] = acc[r];
}

// ---------------------------------------------------------------------------
// Edge scatter: out[dst[e], :] += H[src[e], :]   (one wave per edge)
// float4 gathers (global_load_b128), per-dword f32 atomics into L2.
// ---------------------------------------------------------------------------
__global__ void scatter_add(const float* __restrict__ H, const int* __restrict__ src,
                            const int* __restrict__ dst, float* __restrict__ out,
                            int E, int F) {
  int gw   = (int)(((long long)blockIdx.x * blockDim.x + threadIdx.x) >> 5);
  int lane = threadIdx.x & 31;
  if (gw >= E) return;
  int s = src[gw];
  int d = dst[gw];
  const float4* hp = (const float4*)(H + (size_t)s * F);
  float* op = out + (size_t)d * F;
  for (int j4 = lane; j4 < (F >> 2); j4 += 32) {
    float4 v = hp[j4];
    int j = j4 << 2;
    atomicAdd(op + j + 0, v.x);
    atomicAdd(op + j + 1, v.y);
    atomicAdd(op + j + 2, v.z);
    atomicAdd(op + j + 3, v.w);
  }
}

// h[i,j] = relu(h[i,j]*norm_dst[i] + bias[j]) * norm_src[i]   (in place, x4)
__global__ void finalize_relu_scale(float4* __restrict__ h,
                                    const float* __restrict__ norm_dst,
                                    const float* __restrict__ norm_src,
                                    const float4* __restrict__ bias,
                                    long long total4, int Fq) {
  long long idx4 = (long long)blockIdx.x * blockDim.x + threadIdx.x;
  if (idx4 >= total4) return;
  int i  = (int)(idx4 / Fq);
  int j4 = (int)(idx4 - (long long)i * Fq);
  float4 v = h[idx4];
  float4 bb = bias[j4];
  float nd = norm_dst[i], ns = norm_src[i];
  v.x = fmaxf(v.x * nd + bb.x, 0.f) * ns;
  v.y = fmaxf(v.y * nd + bb.y, 0.f) * ns;
  v.z = fmaxf(v.z * nd + bb.z, 0.f) * ns;
  v.w = fmaxf(v.w * nd + bb.w, 0.f) * ns;
  h[idx4] = v;
}

// out[i,j] = out[i,j]*norm_dst[i] + bias[j]   (in place, x4)
__global__ void finalize_out(float4* __restrict__ out,
                             const float* __restrict__ norm_dst,
                             const float4* __restrict__ bias,
                             long long total4, int Fq) {
  long long idx4 = (long long)blockIdx.x * blockDim.x + threadIdx.x;
  if (idx4 >= total4) return;
  int i  = (int)(idx4 / Fq);
  int j4 = (int)(idx4 - (long long)i * Fq);
  float4 v = out[idx4];
  float4 bb = bias[j4];
  float nd = norm_dst[i];
  v.x = v.x * nd + bb.x;
  v.y = v.y * nd + bb.y;
  v.z = v.z * nd + bb.z;
  v.w = v.w * nd + bb.w;
  out[idx4] = v;
}

// ---------------------------------------------------------------------------
extern "C" void kernel_launch(void* const* d_in, const int* in_sizes, int n_in,
                              void* d_out, int out_size, void* d_ws, size_t ws_size,
                              hipStream_t stream) {
  (void)n_in; (void)out_size; (void)ws_size;
  const float* features = (const float*)d_in[0];   // [N, 512]
  const float* W1       = (const float*)d_in[1];   // [512, 256]
  const float* b1       = (const float*)d_in[2];   // [256]
  const float* W2       = (const float*)d_in[3];   // [256, 128]
  const float* b2       = (const float*)d_in[4];   // [128]
  const int*   src      = (const int*)d_in[5];     // [E]
  const int*   dst      = (const int*)d_in[6];     // [E]
  float*       out      = (float*)d_out;           // [N, 128]

  const int N = in_sizes[0] / GCN_IN_FEATS;        // 50000 (divisible by 16)
  const int E = in_sizes[5];                       // 800000

  // Workspace (floats): [deg_out N][deg_in N][bufA N*256][bufB N*256]
  float* wsf      = (float*)d_ws;
  float* norm_src = wsf;                                 // deg_out -> rsqrt
  float* norm_dst = wsf + N;                             // deg_in  -> rsqrt
  float* bufA     = wsf + 2 * (size_t)N;                 // XW / H1W2 product
  float* bufB     = bufA + (size_t)N * GCN_HIDDEN;       // agg1 -> h1 (scaled)

  const long long nh = (long long)N * GCN_HIDDEN;        // 12.8M
  const long long no = (long long)N * GCN_OUT;           // 6.4M

  // ---- init ----
  zero_f32v4<<<512, 256, 0, stream>>>((float4*)norm_src, (2LL * N) / 4);
  zero_f32v4<<<2048, 256, 0, stream>>>((float4*)bufB, nh / 4);
  zero_f32v4<<<2048, 256, 0, stream>>>((float4*)out, no / 4);

  // ---- degrees -> norms ----
  degree_kernel<<<(E + 255) / 256, 256, 0, stream>>>(src, dst, norm_src, norm_dst, E);
  rsqrt_kernel<<<(2 * N + 255) / 256, 256, 0, stream>>>(norm_src, 2 * N);

  // ---- layer 1: bufA = (X * norm_src) @ W1 ----
  gemm_wmma_lds<GCN_IN_FEATS, GCN_HIDDEN>
      <<<N / 16, 2 * GCN_HIDDEN, 0, stream>>>(features, norm_src, W1, bufA);
  scatter_add<<<(E * 32 + 255) / 256, 256, 0, stream>>>(bufA, src, dst, bufB, E, GCN_HIDDEN);
  finalize_relu_scale<<<(int)((nh / 4 + 255) / 256), 256, 0, stream>>>(
      (float4*)bufB, norm_dst, norm_src, (const float4*)b1, nh / 4, GCN_HIDDEN / 4);

  // ---- layer 2: bufA = h1s @ W2 ----
  gemm_wmma_lds<GCN_HIDDEN, GCN_OUT>
      <<<N / 16, 2 * GCN_OUT, 0, stream>>>(bufB, nullptr, W2, bufA);
  scatter_add<<<(E * 32 + 255) / 256, 256, 0, stream>>>(bufA, src, dst, out, E, GCN_OUT);
  finalize_out<<<(int)((no / 4 + 255) / 256), 256, 0, stream>>>(
      (float4*)out, norm_dst, (const float4*)b2, no / 4, GCN_OUT);
}